// BaseCaps_31894427140204
// MI455X (gfx1250) — compile-verified
//
#include <hip/hip_runtime.h>
#include <math.h>

// EM routing (matrix capsules), N=512, K=288, C=32, P=16, routings=3 (fp32).
// Memory-bound: 5 streaming passes over 302MB votes (~1.5GB @ 23.3TB/s ~ 65us).
// M-step K-reductions are performed with V_WMMA_F32_16X16X4_F32 (fp32 matrix
// core) using B = broadcast rr columns; variance via one-pass E[x^2]-mean^2.

#define N_ 512
#define K_ 288
#define C_ 32
#define P_ 16
#define EPSV 1e-7f

typedef __attribute__((ext_vector_type(2))) float v2f;
typedef __attribute__((ext_vector_type(8))) float v8f;

// ---------------- M-step: one wave per (n,c) pair ----------------
// A layout (16x4 f32): lane m (0-15): VGPR0=A[m,0], VGPR1=A[m,1];
//                      lane m+16:     VGPR0=A[m,2], VGPR1=A[m,3].
// We set A[p, j] = votes[n, k0+2*(lane>>4)+j, c, p]  (p = lane&15),
// B[j, col]     = rr[n, k0+2*(lane>>4)+j, c]   (all 16 cols equal),
// so D[p', col] accumulates sum_k rr*v_p' identically in every column.
// C/D layout: VGPR r -> p' = r + 8*(lane>=16).
template<bool USE_RR, bool LAST>
__global__ __launch_bounds__(128)
void mstep_kernel(const float* __restrict__ votes,
                  const float* __restrict__ beta_a,
                  const float* __restrict__ beta_v,
                  const float* __restrict__ rr,
                  float* __restrict__ ws_means,   // [N,C,16]
                  float* __restrict__ ws_i2v,     // [N,C,16]  1/(2*var)
                  float* __restrict__ ws_lnterm,  // [N,C]     log(act+eps)+prob_main
                  float* __restrict__ out,        // d_out: [N*C act | N*C*16 means]
                  float lambd)
{
    const int wave  = threadIdx.x >> 5;
    const int lane  = threadIdx.x & 31;
    const int idx   = blockIdx.x * 4 + wave;   // n*C + c  (c fastest -> L2 locality)
    const int c     = idx & (C_ - 1);
    const int n     = idx >> 5;
    const int khalf = lane >> 4;               // 0: k0,k1  1: k2,k3
    const int p     = lane & 15;

    const float* vbase = votes + (size_t)n * (K_ * C_ * P_) + (size_t)c * P_ + p;
    const float* rbase = rr    + (size_t)n * (K_ * C_) + c;

    v8f acc1 = {};   // sum_k rr * v
    v8f acc2 = {};   // sum_k rr * v^2
    float rr_s = 0.0f;

    #pragma unroll 2
    for (int k0 = 0; k0 < K_; k0 += 4) {
        const int k = k0 + khalf * 2;
        v2f a, a2, b;
        a.x = vbase[(size_t)k       * (C_ * P_)];
        a.y = vbase[(size_t)(k + 1) * (C_ * P_)];
        if (USE_RR) {
            b.x = rbase[(size_t)k       * C_];
            b.y = rbase[(size_t)(k + 1) * C_];
        } else {
            b.x = 1.0f / (float)C_;
            b.y = 1.0f / (float)C_;
        }
        a2.x = a.x * a.x;
        a2.y = a.y * a.y;
        acc1 = __builtin_amdgcn_wmma_f32_16x16x4_f32(false, a,  false, b, (short)0, acc1, false, false);
        acc2 = __builtin_amdgcn_wmma_f32_16x16x4_f32(false, a2, false, b, (short)0, acc2, false, false);
        rr_s += b.x + b.y;
    }

    // lanes 0-15 summed rr over k%4 in {0,1}; lanes 16-31 over {2,3}.
    const float rr_sum = rr_s + __shfl_xor(rr_s, 16, 32);
    const float inv_rs = 1.0f / rr_sum;

    float means_r[8], i2v_r[8];
    float logsum = 0.0f;
    #pragma unroll
    for (int r = 0; r < 8; ++r) {
        const float m   = acc1[r] * inv_rs;
        const float ex2 = acc2[r] * inv_rs;
        float var = ex2 - m * m;
        var = fmaxf(var, 0.0f);
        const float sd = sqrtf(var);
        const float lg = logf(sd + EPSV);
        logsum += lg;
        means_r[r] = m;
        i2v_r[r]   = 0.5f / var;   // 1/(2*std^2); inf if var==0 (matches ref blowup)
    }
    // sum of log(std+eps) over all 16 poses (halves hold p 0-7 / 8-15)
    const float logsum_t   = logsum + __shfl_xor(logsum, 16, 32);
    const float cost_total = 16.0f * beta_v[c] + rr_sum * logsum_t;
    const float act        = 1.0f / (1.0f + expf(-(lambd * (beta_a[c] - cost_total))));

    const int obase = idx * P_ + khalf * 8;
    if (p == 0) {  // lanes 0 and 16 (values replicated across each half-wave)
        #pragma unroll
        for (int r = 0; r < 8; ++r) {
            ws_means[obase + r] = means_r[r];
            ws_i2v[obase + r]   = i2v_r[r];
            if (LAST) out[N_ * C_ + obase + r] = means_r[r];
        }
        if (lane == 0) {
            ws_lnterm[idx] = logf(act + EPSV) - logsum_t;  // log(act+eps)+prob_main
            if (LAST) out[idx] = act;
        }
    }
}

// ---------------- E-step: one 512-thread block per (k,c); softmax over n ----
__global__ __launch_bounds__(512)
void estep_kernel(const float* __restrict__ votes,
                  const float* __restrict__ ws_means,
                  const float* __restrict__ ws_i2v,
                  const float* __restrict__ ws_lnterm,
                  float* __restrict__ rr)     // [N,K,C]
{
    __shared__ float red[16];
    const int kc   = blockIdx.x;     // k*C + c
    const int c    = kc & (C_ - 1);
    const int k    = kc >> 5;
    const int n    = threadIdx.x;    // 0..511
    const int lane = threadIdx.x & 31;
    const int wv   = threadIdx.x >> 5;

    const float4* vp = (const float4*)(votes   + ((size_t)(n * K_ + k) * C_ + c) * P_);
    const float4* mp = (const float4*)(ws_means + (size_t)(n * C_ + c) * P_);
    const float4* ip = (const float4*)(ws_i2v   + (size_t)(n * C_ + c) * P_);

    float s = 0.0f;
    #pragma unroll
    for (int q = 0; q < 4; ++q) {
        const float4 v = vp[q];
        const float4 m = mp[q];
        const float4 w = ip[q];
        const float dx = v.x - m.x, dy = v.y - m.y, dz = v.z - m.z, dw = v.w - m.w;
        s += dx * dx * w.x + dy * dy * w.y + dz * dz * w.z + dw * dw * w.w;
    }
    const float zz = ws_lnterm[n * C_ + c] - s;   // log(act+eps)+prob_main+prob_exp

    // stable softmax over n (block-wide): max
    float mv = zz;
    #pragma unroll
    for (int off = 16; off > 0; off >>= 1) mv = fmaxf(mv, __shfl_xor(mv, off, 32));
    if (lane == 0) red[wv] = mv;
    __syncthreads();
    float bm = red[0];
    #pragma unroll
    for (int i = 1; i < 16; ++i) bm = fmaxf(bm, red[i]);
    __syncthreads();

    const float e = expf(zz - bm);
    float sv = e;
    #pragma unroll
    for (int off = 16; off > 0; off >>= 1) sv += __shfl_xor(sv, off, 32);
    if (lane == 0) red[wv] = sv;
    __syncthreads();
    float bs = 0.0f;
    #pragma unroll
    for (int i = 0; i < 16; ++i) bs += red[i];

    rr[(size_t)(n * K_ + k) * C_ + c] = e / bs;
}

extern "C" void kernel_launch(void* const* d_in, const int* in_sizes, int n_in,
                              void* d_out, int out_size, void* d_ws, size_t ws_size,
                              hipStream_t stream) {
    // inputs: input_act (unused/dead in reference), votes, beta_a, beta_v, routings(=3)
    const float* votes  = (const float*)d_in[1];
    const float* beta_a = (const float*)d_in[2];
    const float* beta_v = (const float*)d_in[3];
    float* out = (float*)d_out;

    float* ws_means  = (float*)d_ws;                 // N*C*16
    float* ws_i2v    = ws_means  + N_ * C_ * P_;     // N*C*16
    float* ws_lnterm = ws_i2v    + N_ * C_ * P_;     // N*C
    float* ws_rr     = ws_lnterm + N_ * C_;          // N*K*C  (~19MB)

    const dim3 mgrid(N_ * C_ / 4), mblock(128);
    const dim3 egrid(K_ * C_),     eblock(512);

    const float l0 = 0.0f;                                   // 0.01*(1-0.95^0)
    const float l1 = 0.01f * (1.0f - 0.95f);                 // 0.0005
    const float l2 = 0.01f * (1.0f - 0.95f * 0.95f);         // 0.000975

    mstep_kernel<false, false><<<mgrid, mblock, 0, stream>>>(
        votes, beta_a, beta_v, nullptr, ws_means, ws_i2v, ws_lnterm, out, l0);
    estep_kernel<<<egrid, eblock, 0, stream>>>(votes, ws_means, ws_i2v, ws_lnterm, ws_rr);
    mstep_kernel<true, false><<<mgrid, mblock, 0, stream>>>(
        votes, beta_a, beta_v, ws_rr, ws_means, ws_i2v, ws_lnterm, out, l1);
    estep_kernel<<<egrid, eblock, 0, stream>>>(votes, ws_means, ws_i2v, ws_lnterm, ws_rr);
    mstep_kernel<true, true><<<mgrid, mblock, 0, stream>>>(
        votes, beta_a, beta_v, ws_rr, ws_means, ws_i2v, ws_lnterm, out, l2);
}